// ClustCNNEdgeEncoder_88862873354437
// MI455X (gfx1250) — compile-verified
//
#include <hip/hip_runtime.h>

// ClustCNNEdgeEncoder fused gather:
//   out[(i*200 + j) * 5 + c] = (c == 3) ? float(i)
//                                        : data[ clusts[edge_index[side][i]*100 + k] * 5 + c ]
// where j in [0,200), side = j/100, k = j%100.
//
// Bandwidth-bound: 160 MB NT stores to HBM, all gathers L2-resident (data=4MB,
// clusts=0.8MB, edge_index=0.32MB << 192MB L2). No matmul content -> no WMMA;
// CDNA5 path used: async global->LDS staging + s_wait_asynccnt + LDS reads.

#define PTS_PER_CLUST 100
#define ROWS_PER_EDGE 200      // 2 clusters x 100 points
#define FLOATS_PER_EDGE 1000   // 200 rows x 5 cols
#define BLOCK 256

#if __has_builtin(__builtin_amdgcn_global_load_async_to_lds_b32)
#define HAS_ASYNC_LDS 1
#else
#define HAS_ASYNC_LDS 0
#endif

typedef __attribute__((address_space(1))) int* gint_ptr;
typedef __attribute__((address_space(3))) int* lint_ptr;

__global__ __launch_bounds__(BLOCK)
void ClustCNNEdgeEncoder_gather_kernel(const float* __restrict__ data,
                                       const int*   __restrict__ clusts,
                                       const int*   __restrict__ edge_index,
                                       float*       __restrict__ out,
                                       int E) {
  __shared__ int pts[ROWS_PER_EDGE];   // point ids for this edge's 200 rows

  const int i = blockIdx.x;            // edge id
  const int t = threadIdx.x;

  // ---- Stage the 200 point indices into LDS --------------------------------
  if (t < ROWS_PER_EDGE) {
    const int side = t / PTS_PER_CLUST;          // 0: src cluster, 1: dst cluster
    const int k    = t - side * PTS_PER_CLUST;
    const int c    = edge_index[side * E + i];   // uniform per half-block -> scalar-ish
    const int* src = clusts + c * PTS_PER_CLUST + k;
#if HAS_ASYNC_LDS
    // CDNA5 async copy: GLOBAL_LOAD_ASYNC_TO_LDS_B32 (tracked by ASYNCcnt)
    __builtin_amdgcn_global_load_async_to_lds_b32(
        (gint_ptr)(int*)src,             // drop const, generic -> AS1
        (lint_ptr)&pts[t],               // generic -> AS3
        /*offset=*/0, /*cpol=*/0);
#else
    pts[t] = *src;
#endif
  }

#if HAS_ASYNC_LDS
#if __has_builtin(__builtin_amdgcn_s_wait_asynccnt)
  __builtin_amdgcn_s_wait_asynccnt(0);
#else
  asm volatile("s_wait_asynccnt 0" ::: "memory");
#endif
#endif
  __syncthreads();

  // ---- Emit 1000 output dwords for this edge, fully coalesced --------------
  const float fedge = (float)i;
  const size_t base = (size_t)i * FLOATS_PER_EDGE;

  #pragma unroll
  for (int g0 = 0; g0 < FLOATS_PER_EDGE; g0 += BLOCK) {
    const int g = g0 + t;
    if (g < FLOATS_PER_EDGE) {
      const int r = g / 5;             // output row within edge (0..199)
      const int c = g - r * 5;         // column (0..4)
      float v;
      if (c == 3) {
        v = fedge;                     // stamp edge id into column 3
      } else {
        v = data[pts[r] * 5 + c];      // L2-resident random gather
      }
      // write-once streaming output: non-temporal to preserve L2 for gathers
      __builtin_nontemporal_store(v, &out[base + g]);
    }
  }
}

extern "C" void kernel_launch(void* const* d_in, const int* in_sizes, int n_in,
                              void* d_out, int out_size, void* d_ws, size_t ws_size,
                              hipStream_t stream) {
  const float* data       = (const float*)d_in[0];  // [200000, 5] f32
  const int*   clusts     = (const int*)d_in[1];    // [2000, 100] i32
  const int*   edge_index = (const int*)d_in[2];    // [2, 40000]  i32
  float*       out        = (float*)d_out;          // [E*200*5]   f32

  const int E = in_sizes[2] / 2;                    // 40000

  ClustCNNEdgeEncoder_gather_kernel<<<dim3(E), dim3(BLOCK), 0, stream>>>(
      data, clusts, edge_index, out, E);
}